// GNN_27324581937692
// MI455X (gfx1250) — compile-verified
//
#include <hip/hip_runtime.h>
#include <cstdint>
#include <cstddef>

typedef __attribute__((ext_vector_type(2))) float v2f;
typedef __attribute__((ext_vector_type(8))) float v8f;

// ---------------- degree / normalization ----------------

__global__ void __launch_bounds__(256) k_deg_init(float* __restrict__ deg, int N) {
    int i = blockIdx.x * 256 + threadIdx.x;
    if (i < N) deg[i] = 1.0f;  // self-loop contributes 1 to in-degree
}

__global__ void __launch_bounds__(256) k_deg_edges(const long long* __restrict__ col,
                                                   float* __restrict__ deg, int E) {
    int e = blockIdx.x * 256 + threadIdx.x;
    if (e < E) atomicAdd(&deg[(int)col[e]], 1.0f);
}

__global__ void __launch_bounds__(256) k_rsqrt(float* __restrict__ deg, int N) {
    int i = blockIdx.x * 256 + threadIdx.x;
    if (i < N) deg[i] = rsqrtf(deg[i]);  // deg >= 1 always (self-loops)
}

// ---------------- fp32 WMMA GEMM: Y[N,128] = X[N,128] @ W[128,128] ----------------
// One wave computes a 16x128 output strip via 8 accumulators of v_wmma_f32_16x16x4_f32.
// B fragments for the whole 128x128 W are pre-staged in LDS in fragment order:
//   Bfrag[s][n][lane] = { W[kt][c], W[kt+1][c] },  kt = 4s + 2*(lane>>4), c = 16n + (lane&15)
// -> each B fetch is one coalesced, conflict-free ds_load_b64. Total LDS = 64KB.

__global__ void __launch_bounds__(256) k_gemm_wmma(const float* __restrict__ X,
                                                   const float* __restrict__ W,
                                                   float* __restrict__ Y, int N) {
    __shared__ v2f Bfrag[32 * 8 * 32];
    const int tid = threadIdx.x;

#pragma unroll
    for (int i = 0; i < 32; ++i) {
        int entry = tid + 256 * i;          // 0..8191
        int s  = entry >> 8;                // k-step 0..31
        int n  = (entry >> 5) & 7;          // col tile 0..7
        int ln = entry & 31;                // lane
        int kt = 4 * s + 2 * (ln >> 4);
        int c  = 16 * n + (ln & 15);
        v2f val;
        val.x = W[kt * 128 + c];
        val.y = W[(kt + 1) * 128 + c];
        Bfrag[entry] = val;
    }
    __syncthreads();

    const int wave = tid >> 5;
    const int lane = tid & 31;
    const int half = lane >> 4;   // 0: K={4s,4s+1}, 1: K={4s+2,4s+3}
    const int l16  = lane & 15;
    const long rowBase = (long)blockIdx.x * 128 + wave * 16;
    if (rowBase >= N) return;

    const float* xrow = X + (rowBase + l16) * 128;

    v8f acc[8] = {};
    for (int s = 0; s < 32; ++s) {
        const int kt = 4 * s + 2 * half;
        const v2f a = *(const v2f*)(xrow + kt);   // contiguous K-pair, 8B aligned
#pragma unroll
        for (int n = 0; n < 8; ++n) {
            const v2f b = Bfrag[(s * 8 + n) * 32 + lane];
            acc[n] = __builtin_amdgcn_wmma_f32_16x16x4_f32(
                false, a, false, b, (short)0, acc[n], false, false);
        }
    }

    // C/D layout: lane l holds (m = v + 8*(l>>4), n = l&15) in VGPR v
#pragma unroll
    for (int n = 0; n < 8; ++n) {
#pragma unroll
        for (int v = 0; v < 8; ++v) {
            Y[(rowBase + v + 8 * half) * 128 + n * 16 + l16] = acc[n][v];
        }
    }
}

// ---------------- aggregation ----------------

// out[i,f] = xw[i,f]*dinv[i]^2 (+ bias[f])   -- self-loop term, also initializes out
__global__ void __launch_bounds__(256) k_agg_init(const float* __restrict__ xw,
                                                  const float* __restrict__ dinv,
                                                  const float* __restrict__ bias,
                                                  float* __restrict__ out,
                                                  long total, int useBias) {
    long idx = (long)blockIdx.x * 256 + threadIdx.x;
    if (idx >= total) return;
    long i = idx >> 7;
    int  f = (int)(idx & 127);
    float dv = dinv[i];
    float v = xw[idx] * dv * dv;
    if (useBias) v += bias[f];
    out[idx] = v;
}

// per edge: out[col] += xw[row] * dinv[row]*dinv[col]; one wave per edge, float4 per lane
__global__ void __launch_bounds__(256) k_agg_edges(const float* __restrict__ xw,
                                                   const float* __restrict__ dinv,
                                                   const long long* __restrict__ row,
                                                   const long long* __restrict__ col,
                                                   float* __restrict__ out, int E) {
    long t = (long)blockIdx.x * 256 + threadIdx.x;
    int e = (int)(t >> 5);
    if (e >= E) return;
    int lane = (int)(t & 31);
    int r = (int)row[e];
    int c = (int)col[e];
    float wgt = dinv[r] * dinv[c];
    const float4 m = *(const float4*)(xw + (long)r * 128 + lane * 4);
    float* o = out + (long)c * 128 + lane * 4;
    atomicAdd(o + 0, m.x * wgt);
    atomicAdd(o + 1, m.y * wgt);
    atomicAdd(o + 2, m.z * wgt);
    atomicAdd(o + 3, m.w * wgt);
}

// h = relu(h + b1)
__global__ void __launch_bounds__(256) k_bias_relu(float* __restrict__ h,
                                                   const float* __restrict__ bias,
                                                   long total) {
    long idx = (long)blockIdx.x * 256 + threadIdx.x;
    if (idx >= total) return;
    int f = (int)(idx & 127);
    h[idx] = fmaxf(h[idx] + bias[f], 0.0f);
}

// ---------------- launcher ----------------

extern "C" void kernel_launch(void* const* d_in, const int* in_sizes, int n_in,
                              void* d_out, int out_size, void* d_ws, size_t ws_size,
                              hipStream_t stream) {
    const float*     x   = (const float*)d_in[0];
    const long long* ei  = (const long long*)d_in[1];   // int64 edge_index [2,E]
    const float*     W1  = (const float*)d_in[2];
    const float*     b1  = (const float*)d_in[3];
    const float*     W2  = (const float*)d_in[4];
    const float*     b2  = (const float*)d_in[5];
    float* out = (float*)d_out;

    const int N = in_sizes[0] / 128;
    const int E = in_sizes[1] / 2;
    const long long* rowp = ei;
    const long long* colp = ei + E;

    char* ws = (char*)d_ws;
    float* dinv = (float*)ws;
    size_t off = ((size_t)N * sizeof(float) + 255) & ~(size_t)255;
    float* xw = (float*)(ws + off);              // N*128 fp32
    float* h  = xw + (size_t)N * 128;            // N*128 fp32

    const long total = (long)N * 128;
    const int nb_N   = (N + 255) / 256;
    const int nb_E   = (E + 255) / 256;
    const int nb_T   = (int)((total + 255) / 256);
    const int nb_E32 = (int)(((long)E * 32 + 255) / 256);
    const int nb_G   = (N + 127) / 128;

    // normalization: dinv = rsqrt(1 + in-degree)
    k_deg_init <<<nb_N, 256, 0, stream>>>(dinv, N);
    k_deg_edges<<<nb_E, 256, 0, stream>>>(colp, dinv, E);
    k_rsqrt    <<<nb_N, 256, 0, stream>>>(dinv, N);

    // layer 1
    k_gemm_wmma<<<nb_G,   256, 0, stream>>>(x, W1, xw, N);
    k_agg_init <<<nb_T,   256, 0, stream>>>(xw, dinv, b1, h, total, 0);
    k_agg_edges<<<nb_E32, 256, 0, stream>>>(xw, dinv, rowp, colp, h, E);
    k_bias_relu<<<nb_T,   256, 0, stream>>>(h, b1, total);

    // layer 2
    k_gemm_wmma<<<nb_G,   256, 0, stream>>>(h, W2, xw, N);
    k_agg_init <<<nb_T,   256, 0, stream>>>(xw, dinv, b2, out, total, 1);
    k_agg_edges<<<nb_E32, 256, 0, stream>>>(xw, dinv, rowp, colp, out, E);
}